// BodyRecoveryFlow_38328288150113
// MI455X (gfx1250) — compile-verified
//
#include <hip/hip_runtime.h>
#include <hip/hip_bf16.h>
#include <stdint.h>

// ---------------------------------------------------------------------------
// BodyRecoveryFlow for MI455X (gfx1250).
// Bandwidth-bound gather/elementwise pipeline (~0.7-1 GB traffic, ~0.3 GFLOP
// -> ~30-45us HBM roofline @ 23.3 TB/s). No shared-operand GEMM -> WMMA is
// inapplicable; the CDNA5 features that pay are the async global->LDS path
// (15x15 erosion tile staging), b128 vector I/O (4 px/thread), and
// non-temporal stores to keep the 192MB L2 free for the gather-heavy reads.
// ---------------------------------------------------------------------------

typedef float f32x4 __attribute__((ext_vector_type(4)));
typedef float f32x2 __attribute__((ext_vector_type(2)));
typedef int   i32x4 __attribute__((ext_vector_type(4)));

#define Bv   16
#define NFc  13776
#define HWc  (512u * 512u)   // 262144 = 1<<18

// Output offsets (floats), concatenated flat in reference return order.
constexpr size_t O_SYN  = 0;                                  // [B,3,H,W]
constexpr size_t O_GBG  = O_SYN  + (size_t)Bv * 3 * HWc;      // [B,4,H,W]
constexpr size_t O_GSRC = O_GBG  + (size_t)Bv * 4 * HWc;      // [B,6,H,W]
constexpr size_t O_GTSF = O_GSRC + (size_t)Bv * 6 * HWc;      // [B,6,H,W]
constexpr size_t O_T    = O_GTSF + (size_t)Bv * 6 * HWc;      // [B,H,W,2]
constexpr size_t O_BGM  = O_T    + (size_t)Bv * 2 * HWc;      // [2B,1,H,W]
constexpr size_t O_COND = O_BGM  + (size_t)2 * Bv * HWc;      // [B,6,H,W]
constexpr size_t O_J2D  = O_COND + (size_t)Bv * 6 * HWc;      // [B,19,2]

__device__ __forceinline__ void st_nt4(float* p, f32x4 v) {
    __builtin_nontemporal_store(v, (f32x4*)p);
}
__device__ __forceinline__ f32x4 ld4(const float* p) {
    return *(const f32x4*)p;
}

// 6-column background-bit window for 4 pixels: center quad already in cv.
__device__ __forceinline__ void row_bits(const int* row, int w, i32x4 cv, int* e) {
    e[1] = (cv.x == NFc); e[2] = (cv.y == NFc); e[3] = (cv.z == NFc); e[4] = (cv.w == NFc);
    e[0] = (w > 0)       ? (row[w - 1] == NFc) : 1;
    e[5] = (w + 4 < 512) ? (row[w + 4] == NFc) : 1;
}

// ---------------------------------------------------------------------------
// Kernel 1: conds + 3x3 erosions + dependent outputs. 4 pixels per thread;
// the middle erosion row reuses the sf4/rf4 quads (no duplicate b128 load).
// ---------------------------------------------------------------------------
__global__ __launch_bounds__(256)
void cond_crop_kernel(const int* __restrict__ src_fim, const int* __restrict__ ref_fim,
                      const float* __restrict__ table, const float* __restrict__ src_img,
                      float* __restrict__ out)
{
    unsigned t = blockIdx.x * 256u + threadIdx.x;  // quad index in [0, B*HW/4)
    unsigned i = t * 4u;                           // first pixel of the quad
    unsigned b = i >> 18;
    unsigned p = i & (HWc - 1u);
    int h = (int)(p >> 9);
    int w = (int)(p & 511u);                       // multiple of 4, 0..508

    i32x4 sf4 = *(const i32x4*)(src_fim + i);
    i32x4 rf4 = *(const i32x4*)(ref_fim + i);
    int sfA[4] = {sf4.x, sf4.y, sf4.z, sf4.w};
    int rfA[4] = {rf4.x, rf4.y, rf4.z, rf4.w};

    // Per-pixel LUT gathers (3 consecutive floats each).
    float s0[4], s1[4], s2[4], r0[4], r1[4], r2[4];
#pragma unroll
    for (int j = 0; j < 4; ++j) {
        const float* ts = table + 3 * sfA[j];
        const float* tr = table + 3 * rfA[j];
        s0[j] = ts[0]; s1[j] = ts[1]; s2[j] = ts[2];
        r0[j] = tr[0]; r1[j] = tr[1]; r2[j] = tr[2];
    }

    // 3x3 binary erosion of (fim == NF); out-of-image counts as background.
    const int* sfb = src_fim + ((size_t)b << 18);
    const int* rfb = ref_fim + ((size_t)b << 18);
    int sacc[4] = {1, 1, 1, 1}, racc[4] = {1, 1, 1, 1};
    {
        int es[6], er[6];
        // middle row: reuse the quads we already loaded
        row_bits(sfb + (h << 9), w, sf4, es);
        row_bits(rfb + (h << 9), w, rf4, er);
#pragma unroll
        for (int j = 0; j < 4; ++j) {
            sacc[j] &= es[j] & es[j + 1] & es[j + 2];
            racc[j] &= er[j] & er[j + 1] & er[j + 2];
        }
        // top / bottom rows (OOB row is all-background: AND identity)
#pragma unroll
        for (int dy = -1; dy <= 1; dy += 2) {
            int hh = h + dy;
            if ((unsigned)hh >= 512u) continue;
            const int* srow = sfb + (hh << 9);
            const int* rrow = rfb + (hh << 9);
            i32x4 sv = *(const i32x4*)(srow + w);
            i32x4 rv = *(const i32x4*)(rrow + w);
            row_bits(srow, w, sv, es);
            row_bits(rrow, w, rv, er);
#pragma unroll
            for (int j = 0; j < 4; ++j) {
                sacc[j] &= es[j] & es[j + 1] & es[j + 2];
                racc[j] &= er[j] & er[j + 1] & er[j + 2];
            }
        }
    }
    f32x4 scrop = {(float)sacc[0], (float)sacc[1], (float)sacc[2], (float)sacc[3]};
    f32x4 tcrop = {(float)racc[0], (float)racc[1], (float)racc[2], (float)racc[3]};

    f32x4 s0v = {s0[0], s0[1], s0[2], s0[3]};
    f32x4 s1v = {s1[0], s1[1], s1[2], s1[3]};
    f32x4 s2v = {s2[0], s2[1], s2[2], s2[3]};
    f32x4 r0v = {r0[0], r0[1], r0[2], r0[3]};
    f32x4 r1v = {r1[0], r1[1], r1[2], r1[3]};
    f32x4 r2v = {r2[0], r2[1], r2[2], r2[3]};

    // conds = cat(src_cond, ref_cond)
    float* oc = out + O_COND + (size_t)b * 6 * HWc + p;
    st_nt4(oc + 0 * HWc, s0v); st_nt4(oc + 1 * HWc, s1v); st_nt4(oc + 2 * HWc, s2v);
    st_nt4(oc + 3 * HWc, r0v); st_nt4(oc + 4 * HWc, r1v); st_nt4(oc + 5 * HWc, r2v);

    // input_G_src = cat(src_img * (1 - src_crop_mask), src_cond)
    const float* imgb = src_img + (size_t)b * 3 * HWc + p;
    f32x4 inv = (f32x4)(1.0f) - scrop;
    f32x4 i0 = ld4(imgb + 0 * HWc), i1 = ld4(imgb + 1 * HWc), i2 = ld4(imgb + 2 * HWc);
    float* os = out + O_GSRC + (size_t)b * 6 * HWc + p;
    st_nt4(os + 0 * HWc, i0 * inv);
    st_nt4(os + 1 * HWc, i1 * inv);
    st_nt4(os + 2 * HWc, i2 * inv);
    st_nt4(os + 3 * HWc, s0v); st_nt4(os + 4 * HWc, s1v); st_nt4(os + 5 * HWc, s2v);

    // input_G_tsf channels 3..5 = ref_cond
    float* ot = out + O_GTSF + (size_t)b * 6 * HWc + p;
    st_nt4(ot + 3 * HWc, r0v); st_nt4(ot + 4 * HWc, r1v); st_nt4(ot + 5 * HWc, r2v);

    // bg_mask = cat(src_crop_mask, tsf_crop_mask) along batch
    st_nt4(out + O_BGM + ((size_t)b << 18) + p, scrop);
    st_nt4(out + O_BGM + ((size_t)(Bv + b) << 18) + p, tcrop);
}

// ---------------------------------------------------------------------------
// Kernel 2: 15x15 erosion of src background mask + input_G_bg. 128x8 output
// tile per 256-thread block (4 px/thread); 142x22 fim halo tile staged via
// CDNA5 async global->LDS (ASYNCcnt), separable AND in LDS, b128 NT output.
// ---------------------------------------------------------------------------
#define TW 142
#define TH 22

__global__ __launch_bounds__(256)
void bg15_kernel(const int* __restrict__ src_fim, const float* __restrict__ src_img,
                 float* __restrict__ out)
{
    __shared__ int tfim[TH * TW];      // 3124 ints: raw fim halo tile
    __shared__ int rmin[TH][128];      // horizontal AND-of-15 per row

    const int tx = threadIdx.x, ty = threadIdx.y;
    const int tid = ty * 32 + tx;
    const int b = blockIdx.z;
    const int gx0 = blockIdx.x * 128 - 7;
    const int gy0 = blockIdx.y * 8 - 7;
    const int* fimb = src_fim + ((size_t)b << 18);

    // Stage halo tile: in-range via async global->LDS DMA path (ASYNCcnt),
    // out-of-image slots get the background id (erosion pads with bg=1).
    for (int k = tid; k < TH * TW; k += 256) {
        int ly = k / TW;
        int lx = k - ly * TW;
        int gyp = gy0 + ly, gxp = gx0 + lx;
        if ((unsigned)gxp < 512u && (unsigned)gyp < 512u) {
            uint32_t lds_off = (uint32_t)(uintptr_t)(&tfim[k]);
            uint64_t gaddr = (uint64_t)(uintptr_t)(fimb + ((gyp << 9) + gxp));
            asm volatile("global_load_async_to_lds_b32 %0, %1, off"
                         :: "v"(lds_off), "v"(gaddr) : "memory");
        } else {
            tfim[k] = NFc;
        }
    }
    asm volatile("s_wait_asynccnt 0" ::: "memory");
    __syncthreads();

    // Horizontal AND over 15: each thread covers 4 coalesced columns
    // (c = tx + 32q), rows strided by blockDim.y.
    for (int ly = ty; ly < TH; ly += 8) {
        const int* rowp = &tfim[ly * TW];
#pragma unroll
        for (int q = 0; q < 4; ++q) {
            int c = tx + 32 * q;
            int acc = 1;
#pragma unroll
            for (int dx = 0; dx < 15; ++dx)
                acc &= (rowp[c + dx] == NFc) ? 1 : 0;
            rmin[ly][c] = acc;
        }
    }
    __syncthreads();

    // Vertical AND over 15, 4 adjacent columns per thread (b128 LDS reads).
    const int c0 = tx * 4;
    i32x4 m = {1, 1, 1, 1};
#pragma unroll
    for (int dy = 0; dy < 15; ++dy)
        m &= *(const i32x4*)&rmin[ty + dy][c0];
    f32x4 mf = {(float)m.x, (float)m.y, (float)m.z, (float)m.w};

    const int gx = blockIdx.x * 128 + c0;
    const int gy = blockIdx.y * 8 + ty;
    const size_t p = ((size_t)gy << 9) + gx;
    const float* img = src_img + (size_t)b * 3 * HWc + p;
    float* obg = out + O_GBG + (size_t)b * 4 * HWc + p;
    st_nt4(obg + 0 * HWc, ld4(img + 0 * HWc) * mf);
    st_nt4(obg + 1 * HWc, ld4(img + 1 * HWc) * mf);
    st_nt4(obg + 2 * HWc, ld4(img + 2 * HWc) * mf);
    st_nt4(obg + 3 * HWc, mf);
}

// ---------------------------------------------------------------------------
// Kernel 3: flow field T + fused bilinear grid_sample. 4 pixels per thread:
// ref_fim int4, ref_wim 3x b128, T 2x b128 NT, output planes b128 NT. The
// per-pixel triangle/texel gathers stay scalar (inherently random access).
// ---------------------------------------------------------------------------
__global__ __launch_bounds__(256)
void tsf_kernel(const int* __restrict__ ref_fim, const float* __restrict__ ref_wim,
                const float* __restrict__ f2verts, const float* __restrict__ src_img,
                float* __restrict__ out)
{
    unsigned t = blockIdx.x * 256u + threadIdx.x;
    unsigned i = t * 4u;
    unsigned b = i >> 18;
    unsigned p = i & (HWc - 1u);

    i32x4 rf4 = *(const i32x4*)(ref_fim + i);
    int rfA[4] = {rf4.x, rf4.y, rf4.z, rf4.w};

    const float* wb = ref_wim + (size_t)i * 3;     // 12 consecutive floats
    f32x4 wv0 = ld4(wb + 0), wv1 = ld4(wb + 4), wv2 = ld4(wb + 8);
    float wA[12] = {wv0.x, wv0.y, wv0.z, wv0.w,
                    wv1.x, wv1.y, wv1.z, wv1.w,
                    wv2.x, wv2.y, wv2.z, wv2.w};

    float Tx[4], Ty[4];
#pragma unroll
    for (int j = 0; j < 4; ++j) {
        int rf = rfA[j];
        if (rf == NFc) {
            Tx[j] = -2.0f; Ty[j] = -2.0f;          // background pixel
        } else {
            const float* tv = f2verts + ((size_t)b * NFc + (size_t)rf) * 6;
            f32x2 v0 = *(const f32x2*)(tv + 0);
            f32x2 v1 = *(const f32x2*)(tv + 2);
            f32x2 v2 = *(const f32x2*)(tv + 4);
            float w0 = wA[3 * j], w1 = wA[3 * j + 1], w2 = wA[3 * j + 2];
            Tx[j] =  (w0 * v0.x + w1 * v1.x + w2 * v2.x);
            Ty[j] = -(w0 * v0.y + w1 * v1.y + w2 * v2.y);   // y-flip folded in
        }
    }
    float* oT = out + O_T + (size_t)i * 2;
    st_nt4(oT + 0, (f32x4){Tx[0], Ty[0], Tx[1], Ty[1]});
    st_nt4(oT + 4, (f32x4){Tx[2], Ty[2], Tx[3], Ty[3]});

    // Bilinear sampling (align_corners=True, zero padding).
    const float* ib = src_img + (size_t)b * 3 * HWc;
    float c0[4], c1[4], c2[4];
#pragma unroll
    for (int j = 0; j < 4; ++j) {
        float gx = (Tx[j] + 1.0f) * 0.5f * 511.0f;
        float gy = (Ty[j] + 1.0f) * 0.5f * 511.0f;
        float x0f = floorf(gx), y0f = floorf(gy);
        float wx = gx - x0f, wy = gy - y0f;
        int x0 = (int)x0f, y0 = (int)y0f;
        float a0 = 0.0f, a1 = 0.0f, a2 = 0.0f;
#pragma unroll
        for (int dy = 0; dy < 2; ++dy) {
            int yi = y0 + dy;
            if ((unsigned)yi >= 512u) continue;
            float wyv = dy ? wy : (1.0f - wy);
#pragma unroll
            for (int dx = 0; dx < 2; ++dx) {
                int xi = x0 + dx;
                if ((unsigned)xi >= 512u) continue;
                float wgt = (dx ? wx : (1.0f - wx)) * wyv;
                size_t idx = ((size_t)yi << 9) + xi;
                a0 += ib[0 * HWc + idx] * wgt;
                a1 += ib[1 * HWc + idx] * wgt;
                a2 += ib[2 * HWc + idx] * wgt;
            }
        }
        c0[j] = a0; c1[j] = a1; c2[j] = a2;
    }

    f32x4 s0 = {c0[0], c0[1], c0[2], c0[3]};
    f32x4 s1 = {c1[0], c1[1], c1[2], c1[3]};
    f32x4 s2 = {c2[0], c2[1], c2[2], c2[3]};
    float* osyn = out + O_SYN + (size_t)b * 3 * HWc + p;
    st_nt4(osyn + 0 * HWc, s0); st_nt4(osyn + 1 * HWc, s1); st_nt4(osyn + 2 * HWc, s2);
    float* ot = out + O_GTSF + (size_t)b * 6 * HWc + p;
    st_nt4(ot + 0 * HWc, s0); st_nt4(ot + 1 * HWc, s1); st_nt4(ot + 2 * HWc, s2);
}

// ---------------------------------------------------------------------------
// Kernel 4: ref_j2d passthrough (608 floats = 152 float4).
// ---------------------------------------------------------------------------
__global__ void j2d_kernel(const float* __restrict__ j2d, float* __restrict__ out)
{
    int i = threadIdx.x;
    if (i < (Bv * 19 * 2) / 4) {
        f32x4 v = *(const f32x4*)(j2d + i * 4);
        st_nt4(out + O_J2D + i * 4, v);
    }
}

// ---------------------------------------------------------------------------
extern "C" void kernel_launch(void* const* d_in, const int* in_sizes, int n_in,
                              void* d_out, int out_size, void* d_ws, size_t ws_size,
                              hipStream_t stream)
{
    (void)in_sizes; (void)n_in; (void)out_size; (void)d_ws; (void)ws_size;

    const float* src_img = (const float*)d_in[0];   // [B,3,H,W]
    const float* f2verts = (const float*)d_in[1];   // [B,NF,3,2]
    const float* ref_wim = (const float*)d_in[2];   // [B,H,W,3]
    const float* table   = (const float*)d_in[3];   // [NF+1,3]
    const float* ref_j2d = (const float*)d_in[4];   // [B,19,2]
    const int*   src_fim = (const int*)d_in[5];     // [B,H,W]
    const int*   ref_fim = (const int*)d_in[6];     // [B,H,W]
    float* out = (float*)d_out;

    const int nquad_blocks = (int)((Bv * HWc) / 4u / 256u);   // 4096

    cond_crop_kernel<<<nquad_blocks, 256, 0, stream>>>(src_fim, ref_fim, table, src_img, out);

    dim3 bgrid(512 / 128, 512 / 8, Bv);   // 128x8 output tile per block
    dim3 bblk(32, 8);
    bg15_kernel<<<bgrid, bblk, 0, stream>>>(src_fim, src_img, out);

    tsf_kernel<<<nquad_blocks, 256, 0, stream>>>(ref_fim, ref_wim, f2verts, src_img, out);

    j2d_kernel<<<1, 256, 0, stream>>>(ref_j2d, out);
}